// WindowedValuesMetadataAttention_5600637354234
// MI455X (gfx1250) — compile-verified
//
#include <hip/hip_runtime.h>
#include <hip/hip_bf16.h>
#include <math.h>

// ---------------- CDNA5 WMMA fp32 16x16x4 helpers ----------------
typedef __attribute__((ext_vector_type(2))) float v2f;
typedef __attribute__((ext_vector_type(8))) float v8f;

#define SCALE_F 0.17677669529663687f   // 1/sqrt(32)

__device__ __forceinline__ v8f wmma4(v2f a, v2f b, v8f c) {
  // D = A(16x4 f32) * B(4x16 f32) + C(16x16 f32)
  return __builtin_amdgcn_wmma_f32_16x16x4_f32(
      /*neg_a=*/false, a, /*neg_b=*/false, b,
      /*c_mod=*/(short)0, c, /*reuse_a=*/false, /*reuse_b=*/false);
}

__device__ __forceinline__ float clip5(float x) {
  return fminf(fmaxf(x, -5.0f), 5.0f);
}

// ---------------- K0: argsort(dt) + window source map ----------------
// sorted_idx[b][rank] = original source index; stable ascending.
// in_map[g][sw] = original source feeding slot sw of window g (sort + roll(+2) fused).
__global__ void sort_kernel(const float* __restrict__ dt,
                            int* __restrict__ sorted_idx,
                            int* __restrict__ in_map) {
  int tid = threadIdx.x;
  if (tid < 64) {
    int b = tid >> 3, j = tid & 7;
    float x = dt[b * 8 + j];
    int rank = 0;
    for (int i = 0; i < 8; ++i) {
      float y = dt[b * 8 + i];
      if (y < x || (y == x && i < j)) rank++;
    }
    sorted_idx[b * 8 + rank] = j;
  }
  __syncthreads();
  if (tid < 64) {
    int g = tid >> 2, sw = tid & 3;     // g in [0,16), sw in [0,4)
    int b = g >> 1, w = g & 1;
    int p = w * 4 + sw;                 // rolled position
    int sp = (p + 6) & 7;               // (p - 2) mod 8  -> sorted position
    in_map[g * 4 + sw] = sorted_idx[b * 8 + sp];
  }
}

// ---------------- K1: qkv projection ----------------
// C(16384x768) = values(16384x256) @ W_qkv(256x768); each wave computes a
// 64x16 tile (4 M-tiles share one B fragment per k-slice). Clip cols<512.
// Scatter to head-major [b][s][h][t][32] layout for Qp / Kp / V.
__global__ __launch_bounds__(256) void proj_qkv_kernel(
    const float* __restrict__ vals, const float* __restrict__ Wqkv,
    float* __restrict__ Qp, float* __restrict__ Kp, float* __restrict__ Vv) {
  int wv = threadIdx.x >> 5;
  int lane = threadIdx.x & 31;
  int ln = lane & 15, hi = lane >> 4;
  int tg = blockIdx.x * 8 + wv;         // 12288 wave-tiles (256 Mgrp x 48 N)
  int mg = tg / 48, nt = tg % 48;
  int m0 = mg * 64, n0 = nt * 16;

  const float* arow0 = vals + (m0 + ln) * 256;
  const float* arow1 = arow0 + 16 * 256;
  const float* arow2 = arow0 + 32 * 256;
  const float* arow3 = arow0 + 48 * 256;
  __builtin_prefetch(arow0, 0, 3);
  __builtin_prefetch(arow1, 0, 3);
  __builtin_prefetch(arow2, 0, 3);
  __builtin_prefetch(arow3, 0, 3);

  v8f acc0 = {}, acc1 = {}, acc2 = {}, acc3 = {};
#pragma unroll 4
  for (int kk = 0; kk < 64; ++kk) {
    int d = kk * 4 + 2 * hi;
    v2f b;
    b.x = Wqkv[d * 768 + n0 + ln];
    b.y = Wqkv[(d + 1) * 768 + n0 + ln];
    v2f a0, a1, a2, a3;
    a0.x = arow0[d]; a0.y = arow0[d + 1];
    a1.x = arow1[d]; a1.y = arow1[d + 1];
    a2.x = arow2[d]; a2.y = arow2[d + 1];
    a3.x = arow3[d]; a3.y = arow3[d + 1];
    acc0 = wmma4(a0, b, acc0);
    acc1 = wmma4(a1, b, acc1);
    acc2 = wmma4(a2, b, acc2);
    acc3 = wmma4(a3, b, acc3);
  }

  // region is wave-uniform: a 16-wide column tile never crosses q/k/v bounds
  int region = __builtin_amdgcn_readfirstlane(n0 >> 8);  // 0=q 1=k 2=v
  float* dst = (region == 0) ? Qp : (region == 1) ? Kp : Vv;
  bool doclip = region < 2;
  int cc = (n0 & 255) + ln;
  int h_ = cc >> 5, dh = cc & 31;
  v8f accs[4] = {acc0, acc1, acc2, acc3};
#pragma unroll
  for (int t = 0; t < 4; ++t) {
#pragma unroll
    for (int v = 0; v < 8; ++v) {
      int m = m0 + t * 16 + v + 8 * hi;  // global row
      float x = accs[t][v];
      if (doclip) x = clip5(x);
      int b_ = m >> 11, s_ = (m >> 8) & 7, t_ = m & 255;
      dst[((((b_ * 8 + s_) * 8 + h_) * 256 + t_) << 5) + dh] = x;
    }
  }
}

// ---------------- K2: meta projection ----------------
// C(16384x512) = meta(16384x64) @ W_meta(64x512); 64x16 tile per wave;
// clip all; scatter Qm / Km.
__global__ __launch_bounds__(256) void proj_meta_kernel(
    const float* __restrict__ meta, const float* __restrict__ Wm,
    float* __restrict__ Qm, float* __restrict__ Km) {
  int wv = threadIdx.x >> 5;
  int lane = threadIdx.x & 31;
  int ln = lane & 15, hi = lane >> 4;
  int tg = blockIdx.x * 8 + wv;         // 8192 wave-tiles (256 Mgrp x 32 N)
  int mg = tg / 32, nt = tg % 32;
  int m0 = mg * 64, n0 = nt * 16;

  const float* arow0 = meta + (m0 + ln) * 64;
  const float* arow1 = arow0 + 16 * 64;
  const float* arow2 = arow0 + 32 * 64;
  const float* arow3 = arow0 + 48 * 64;
  __builtin_prefetch(arow0, 0, 3);
  __builtin_prefetch(arow2, 0, 3);

  v8f acc0 = {}, acc1 = {}, acc2 = {}, acc3 = {};
#pragma unroll
  for (int kk = 0; kk < 16; ++kk) {
    int d = kk * 4 + 2 * hi;
    v2f b;
    b.x = Wm[d * 512 + n0 + ln];
    b.y = Wm[(d + 1) * 512 + n0 + ln];
    v2f a0, a1, a2, a3;
    a0.x = arow0[d]; a0.y = arow0[d + 1];
    a1.x = arow1[d]; a1.y = arow1[d + 1];
    a2.x = arow2[d]; a2.y = arow2[d + 1];
    a3.x = arow3[d]; a3.y = arow3[d + 1];
    acc0 = wmma4(a0, b, acc0);
    acc1 = wmma4(a1, b, acc1);
    acc2 = wmma4(a2, b, acc2);
    acc3 = wmma4(a3, b, acc3);
  }

  int region = __builtin_amdgcn_readfirstlane(n0 >> 8);  // 0=qm 1=km
  float* dst = (region == 0) ? Qm : Km;
  int cc = (n0 & 255) + ln;
  int h_ = cc >> 5, dh = cc & 31;
  v8f accs[4] = {acc0, acc1, acc2, acc3};
#pragma unroll
  for (int t = 0; t < 4; ++t) {
#pragma unroll
    for (int v = 0; v < 8; ++v) {
      int m = m0 + t * 16 + v + 8 * hi;
      float x = clip5(accs[t][v]);
      int b_ = m >> 11, s_ = (m >> 8) & 7, t_ = m & 255;
      dst[((((b_ * 8 + s_) * 8 + h_) * 256 + t_) << 5) + dh] = x;
    }
  }
}

// ---------------- K3: fused windowed attention ----------------
// One block = one (window g, head h, 16-row q-tile). 4 waves, 128 threads.
// Scores (16 x 1024) held in LDS, softmax via wave32 shuffles, then P @ V.
__global__ __launch_bounds__(128) void attn_kernel(
    const float* __restrict__ Qp, const float* __restrict__ Kp,
    const float* __restrict__ Vv, const float* __restrict__ Qm,
    const float* __restrict__ Km, const unsigned char* __restrict__ mask,
    const int* __restrict__ in_map, float* __restrict__ AO) {
  __shared__ float sc[16 * 1024];       // 64 KB score tile (reused for reduce)

  int wv = threadIdx.x >> 5;
  int lane = threadIdx.x & 31;
  int ln = lane & 15, hi = lane >> 4;

  int blk = blockIdx.x;                 // 8192 = 16 g * 8 h * 64 qt
  int qt = blk & 63;
  int gh = blk >> 6;
  int h = gh & 7;
  int g = gh >> 3;
  int b_ = g >> 1;

  int smap[4];
#pragma unroll
  for (int i = 0; i < 4; ++i) smap[i] = in_map[g * 4 + i];

  // ---- load Q fragments (A-matrix, 8 k-slices of 4 for Qp and Qm) ----
  int mq = qt * 16 + ln;                // q row (token l) for this lane
  int sq = smap[mq >> 8], tq = mq & 255;
  int rowQ = ((((b_ * 8 + sq) * 8 + h) * 256 + tq) << 5);
  v2f aQp[8], aQm[8];
#pragma unroll
  for (int kk = 0; kk < 8; ++kk) {
    int d = kk * 4 + 2 * hi;
    aQp[kk].x = Qp[rowQ + d];     aQp[kk].y = Qp[rowQ + d + 1];
    aQm[kk].x = Qm[rowQ + d];     aQm[kk].y = Qm[rowQ + d + 1];
  }

  // ---- phase A: score tiles (each wave owns distinct 16 columns per kt) ----
  for (int kt = 0; kt < 16; ++kt) {
    int jcol = (kt * 4 + wv) * 16 + ln; // key token l in [0,1024)
    int sj = smap[jcol >> 8], tj = jcol & 255;
    int rowK = ((((b_ * 8 + sj) * 8 + h) * 256 + tj) << 5);
    __builtin_prefetch(&Kp[rowK], 0, 3);
    __builtin_prefetch(&Km[rowK], 0, 3);
    v8f acc = {};
#pragma unroll
    for (int kk = 0; kk < 8; ++kk) {
      int d = kk * 4 + 2 * hi;
      v2f bK, bM;
      bK.x = Kp[rowK + d];  bK.y = Kp[rowK + d + 1];
      acc = wmma4(aQp[kk], bK, acc);
      bM.x = Km[rowK + d];  bM.y = Km[rowK + d + 1];
      acc = wmma4(aQm[kk], bM, acc);
    }
    bool mk = mask[(b_ * 8 + sj) * 256 + tj] != 0;
#pragma unroll
    for (int v = 0; v < 8; ++v) {
      float x = acc[v] * SCALE_F;
      if (mk) x = -__builtin_inff();
      sc[(v + 8 * hi) * 1024 + jcol] = x;
    }
  }
  __syncthreads();

  // ---- phase B: softmax over each of 16 rows (8 lanes per row) ----
  {
    int row = threadIdx.x >> 3;
    int sub = threadIdx.x & 7;
    float* r = &sc[row * 1024];
    float mx = -__builtin_inff();
    for (int c = sub; c < 1024; c += 8) mx = fmaxf(mx, r[c]);
#pragma unroll
    for (int o = 1; o < 8; o <<= 1) mx = fmaxf(mx, __shfl_xor(mx, o, 32));
    float sum = 0.f;
    for (int c = sub; c < 1024; c += 8) {
      float e = expf(r[c] - mx);
      r[c] = e;
      sum += e;
    }
#pragma unroll
    for (int o = 1; o < 8; o <<= 1) sum += __shfl_xor(sum, o, 32);
    float inv = 1.0f / sum;
    for (int c = sub; c < 1024; c += 8) r[c] *= inv;
  }
  __syncthreads();

  // ---- phase C: out(16x32) = P(16x1024) @ V(1024x32) ----
  // wave pairs split the K range; n-tile = wv&1, k-half = wv>>1.
  int n0 = (wv & 1) * 16;
  int kh = wv >> 1;
  v8f acc = {};
  for (int i = 0; i < 128; ++i) {
    int k0 = kh * 512 + i * 4;
    v2f a, bv;
    a.x = sc[ln * 1024 + k0 + 2 * hi];
    a.y = sc[ln * 1024 + k0 + 2 * hi + 1];
    int kd0 = k0 + 2 * hi;
    int sv0 = smap[kd0 >> 8], tv0 = kd0 & 255;
    int kd1 = kd0 + 1;
    int sv1 = smap[kd1 >> 8], tv1 = kd1 & 255;
    bv.x = Vv[((((b_ * 8 + sv0) * 8 + h) * 256 + tv0) << 5) + n0 + ln];
    bv.y = Vv[((((b_ * 8 + sv1) * 8 + h) * 256 + tv1) << 5) + n0 + ln];
    acc = wmma4(a, bv, acc);
  }
  __syncthreads();                      // score buffer now dead -> reuse
  if (wv >= 2) {
    float* p = &sc[(wv - 2) * 256];
#pragma unroll
    for (int v = 0; v < 8; ++v) p[(v + 8 * hi) * 16 + ln] = acc[v];
  }
  __syncthreads();
  if (wv < 2) {
    const float* p = &sc[wv * 256];
#pragma unroll
    for (int v = 0; v < 8; ++v) {
      int m = v + 8 * hi;
      float x = acc[v] + p[m * 16 + ln];
      int l = qt * 16 + m;
      AO[(g * 1024 + l) * 256 + h * 32 + n0 + ln] = x;
    }
  }
}

// ---------------- K4: output projection + roll(-2) + unsort scatter ----------------
__global__ __launch_bounds__(256) void out_proj_kernel(
    const float* __restrict__ AO, const float* __restrict__ Wout,
    const float* __restrict__ bout, const int* __restrict__ sorted_idx,
    float* __restrict__ out) {
  int wv = threadIdx.x >> 5;
  int lane = threadIdx.x & 31;
  int ln = lane & 15, hi = lane >> 4;
  int tg = blockIdx.x * 8 + wv;         // 4096 wave-tiles (256 Mgrp x 16 N)
  int mg = tg >> 4, nt = tg & 15;
  int m0 = mg * 64, n0 = nt * 16;

  const float* arow0 = AO + (m0 + ln) * 256;
  const float* arow1 = arow0 + 16 * 256;
  const float* arow2 = arow0 + 32 * 256;
  const float* arow3 = arow0 + 48 * 256;
  __builtin_prefetch(arow0, 0, 3);
  __builtin_prefetch(arow1, 0, 3);
  __builtin_prefetch(arow2, 0, 3);
  __builtin_prefetch(arow3, 0, 3);

  v8f acc0 = {}, acc1 = {}, acc2 = {}, acc3 = {};
#pragma unroll 4
  for (int kk = 0; kk < 64; ++kk) {
    int d = kk * 4 + 2 * hi;
    v2f b;
    b.x = Wout[d * 256 + n0 + ln];
    b.y = Wout[(d + 1) * 256 + n0 + ln];
    v2f a0, a1, a2, a3;
    a0.x = arow0[d]; a0.y = arow0[d + 1];
    a1.x = arow1[d]; a1.y = arow1[d + 1];
    a2.x = arow2[d]; a2.y = arow2[d + 1];
    a3.x = arow3[d]; a3.y = arow3[d + 1];
    acc0 = wmma4(a0, b, acc0);
    acc1 = wmma4(a1, b, acc1);
    acc2 = wmma4(a2, b, acc2);
    acc3 = wmma4(a3, b, acc3);
  }

  int n = n0 + ln;
  float bias = bout[n];
  v8f accs[4] = {acc0, acc1, acc2, acc3};
#pragma unroll
  for (int t = 0; t < 4; ++t) {
#pragma unroll
    for (int v = 0; v < 8; ++v) {
      int m = m0 + t * 16 + v + 8 * hi; // attention row = g*1024 + l
      int g = m >> 10, l = m & 1023;
      int lp = (l + 1022) & 1023;       // (l - 2) mod 1024  (output roll)
      int b_ = g >> 1;
      int sp = (g & 1) * 4 + (lp >> 8); // sorted position
      int t_ = lp & 255;
      int j = sorted_idx[b_ * 8 + sp];  // original source index
      out[(((b_ * 8 + j) * 256 + t_) << 8) + n] = accs[t][v] + bias;
    }
  }
}

// ---------------- launcher ----------------
extern "C" void kernel_launch(void* const* d_in, const int* in_sizes, int n_in,
                              void* d_out, int out_size, void* d_ws, size_t ws_size,
                              hipStream_t stream) {
  const float* dt    = (const float*)d_in[0];
  const float* vals  = (const float*)d_in[1];
  const float* meta  = (const float*)d_in[2];
  const unsigned char* mask = (const unsigned char*)d_in[3];
  const float* Wqkv  = (const float*)d_in[4];
  const float* Wmeta = (const float*)d_in[5];
  const float* Wout  = (const float*)d_in[6];
  const float* bout  = (const float*)d_in[7];
  float* out = (float*)d_out;

  const size_t NPROJ = (size_t)16384 * 256;   // per-tensor scratch (16 MB)
  float* ws = (float*)d_ws;
  float* Qp = ws;
  float* Kp = Qp + NPROJ;
  float* Vv = Kp + NPROJ;
  float* Qm = Vv + NPROJ;
  float* Km = Qm + NPROJ;
  float* AO = Km + NPROJ;
  int* sidx = (int*)(AO + NPROJ);
  int* imap = sidx + 64;

  sort_kernel<<<1, 64, 0, stream>>>(dt, sidx, imap);
  proj_qkv_kernel<<<1536, 256, 0, stream>>>(vals, Wqkv, Qp, Kp, Vv);
  proj_meta_kernel<<<1024, 256, 0, stream>>>(meta, Wmeta, Qm, Km);
  attn_kernel<<<8192, 128, 0, stream>>>(Qp, Kp, Vv, Qm, Km, mask, imap, AO);
  out_proj_kernel<<<512, 256, 0, stream>>>(AO, Wout, bout, sidx, out);
}